// RGCN_Network_6451040878730
// MI455X (gfx1250) — compile-verified
//
#include <hip/hip_runtime.h>

#define D     256
#define S     2048
#define BATCH 8
#define BM    128
#define BN    128
#define KC    64
#define AS_STRIDE 68   // words; 272B, 16B-aligned rows, conflict-free wmma reads
#define BT_STRIDE 68   // transposed B tile: [n][k], K contiguous

typedef __attribute__((ext_vector_type(2))) float v2f;
typedef __attribute__((ext_vector_type(4))) float v4f;
typedef __attribute__((ext_vector_type(8))) float v8f;

// ---------------- Kernel 1: partial per-batch column sums over S -------------
__global__ void colsum_partial(const float* __restrict__ x, float* __restrict__ partial) {
    int b     = blockIdx.x >> 3;
    int chunk = blockIdx.x & 7;
    int d     = threadIdx.x;
    const float* p = x + (b * S + chunk * 256) * D + d;
    float acc = 0.f;
#pragma unroll 8
    for (int s = 0; s < 256; ++s) acc += p[s * D];
    partial[blockIdx.x * D + d] = acc;
}

// ------- Kernel 2: reduce partials -> sums; build per-t stencil coeffs ------
// u_j[b,t,:] = sum_{o=0..4} w[t][j][o] * x[b, t-1+o, :]  +  w[t][j][5] * sums[b,:]
__global__ void reduce_and_coef(const float* __restrict__ partial,
                                const float* __restrict__ comp,
                                float* __restrict__ sums,
                                float* __restrict__ coef) {
    int idx = blockIdx.x * 256 + threadIdx.x;   // 0..2047
    int b = idx >> 8, d = idx & 255;
    float s = 0.f;
#pragma unroll
    for (int c = 0; c < 8; ++c) s += partial[(b * 8 + c) * D + d];
    sums[idx] = s;

    int t = idx;
    float inv1 = 1.0f / (float)(1 + (t + 2 < S));         // r=1: {t, t+2}
    int   d2   = (t >= 1) + (t + 3 < S);                  // r=2: {t-1, t+3}
    float inv2 = d2 ? 1.0f / (float)d2 : 0.f;
    int   bc   = (t >= 1) + 1 + (t + 1 < S) + (t + 2 < S) + (t + 3 < S);
    float inv3 = 1.0f / (float)(S - bc);                  // r=3: complement of band
#pragma unroll
    for (int j = 0; j < 2; ++j) {
        float c0 = comp[0 * 2 + j], c1 = comp[1 * 2 + j];
        float c2 = comp[2 * 2 + j], c3 = comp[3 * 2 + j];
        float m3 = c3 * inv3;
        float* w = coef + t * 12 + j * 6;
        w[0] = (t >= 1)    ? (c2 * inv2 - m3) : 0.f;      // x[t-1]
        w[1] =               c1 * inv1 - m3;              // x[t]
        w[2] = (t + 1 < S) ? (c0        - m3) : 0.f;      // x[t+1]
        w[3] = (t + 2 < S) ? (c1 * inv1 - m3) : 0.f;      // x[t+2]
        w[4] = (t + 3 < S) ? (c2 * inv2 - m3) : 0.f;      // x[t+3]
        w[5] = m3;                                        // * sums[b]
    }
}

// --------- Kernel 3: fused WMMA GEMM  out = [u0|u1|x] @ [B0;B1;root] + bias --
__global__ __launch_bounds__(256) void rgcn_gemm(const float* __restrict__ x,
                                                 const float* __restrict__ basis,
                                                 const float* __restrict__ root,
                                                 const float* __restrict__ bias,
                                                 const float* __restrict__ sums,
                                                 const float* __restrict__ coef,
                                                 float* __restrict__ out) {
    __shared__ float As[BM * AS_STRIDE];    // [row][k]   128x64 (+pad)
    __shared__ float Bst[BN * BT_STRIDE];   // [n][k]     128x64 (+pad), transposed

    const int tid      = threadIdx.x;
    const int rowBase  = blockIdx.x * BM;
    const int b        = rowBase / S;
    const int tBase    = rowBase % S;
    const int colBase0 = blockIdx.y * BN;

    const int wid     = tid >> 5;
    const int lane    = tid & 31;
    const int waveRow = wid & 3;        // 4 row groups of 32 rows
    const int waveCol = wid >> 2;       // 2 col groups of 64 cols
    const int lmod    = lane & 15;
    const int lhi     = lane >> 4;
    const int koff    = lhi * 2;        // per-ISA f32 A/B K-pair selector
    const int mrow0   = waveRow * 32 + lmod;
    const int mrow1   = mrow0 + 16;

    v8f acc[2][4] = {};                 // 2 row-tiles x 4 col-tiles of 16x16

#pragma unroll 1
    for (int kc = 0; kc < 12; ++kc) {
        const int seg    = kc >> 2;          // 0:u0  1:u1  2:x
        const int kcBase = (kc & 3) * KC;

        // ---- stage A tile (128 x 64) in float4 quads ----
        if (seg == 2) {
#pragma unroll
            for (int e = 0; e < 8; ++e) {
                int idx = e * 256 + tid;
                int c4 = (idx & 15) * 4;
                int r  = idx >> 4;
                *(v4f*)&As[r * AS_STRIDE + c4] =
                    *(const v4f*)&x[(b * S + tBase + r) * D + kcBase + c4];
            }
        } else {
#pragma unroll 2
            for (int e = 0; e < 8; ++e) {
                int idx = e * 256 + tid;
                int c4 = (idx & 15) * 4;
                int r  = idx >> 4;
                int t  = tBase + r;
                int kd = kcBase + c4;
                const float* w = coef + t * 12 + seg * 6;
                v4f val = w[5] * (*(const v4f*)&sums[b * D + kd]);
#pragma unroll
                for (int o = 0; o < 5; ++o) {
                    int sr = t - 1 + o;
                    sr = sr < 0 ? 0 : (sr >= S ? S - 1 : sr);   // coeff==0 when OOB
                    val += w[o] * (*(const v4f*)&x[(b * S + sr) * D + kd]);
                }
                *(v4f*)&As[r * AS_STRIDE + c4] = val;
            }
        }

        // ---- stage B tile transposed: Bst[n][k] (64K x 128N) ----
        const float* Wseg = (seg == 0) ? basis : (seg == 1 ? basis + D * D : root);
#pragma unroll
        for (int e = 0; e < 8; ++e) {
            int idx = e * 256 + tid;
            int n4 = (idx & 31) * 4;
            int kr = idx >> 5;                     // 0..63
            v4f wv = *(const v4f*)&Wseg[(kcBase + kr) * D + colBase0 + n4];
            Bst[(n4 + 0) * BT_STRIDE + kr] = wv.x;
            Bst[(n4 + 1) * BT_STRIDE + kr] = wv.y;
            Bst[(n4 + 2) * BT_STRIDE + kr] = wv.z;
            Bst[(n4 + 3) * BT_STRIDE + kr] = wv.w;
        }
        __syncthreads();

        // ---- compute: 16 K-steps, 8 V_WMMA_F32_16X16X4_F32 each ----
#pragma unroll
        for (int kk = 0; kk < KC; kk += 4) {
            v2f a0 = *(const v2f*)&As[mrow0 * AS_STRIDE + kk + koff];
            v2f a1 = *(const v2f*)&As[mrow1 * AS_STRIDE + kk + koff];
#pragma unroll
            for (int sub = 0; sub < 4; ++sub) {
                int ncol = waveCol * 64 + sub * 16 + lmod;
                v2f bv = *(const v2f*)&Bst[ncol * BT_STRIDE + kk + koff];
                acc[0][sub] = __builtin_amdgcn_wmma_f32_16x16x4_f32(
                    false, a0, false, bv, (short)0, acc[0][sub], false, false);
                acc[1][sub] = __builtin_amdgcn_wmma_f32_16x16x4_f32(
                    false, a1, false, bv, (short)0, acc[1][sub], false, false);
            }
        }
        __syncthreads();
    }

    // ---- store with bias (C/D layout: VGPR j -> M = j + 8*lhi, N = lmod) ----
#pragma unroll
    for (int rt = 0; rt < 2; ++rt) {
#pragma unroll
        for (int sub = 0; sub < 4; ++sub) {
            int ncol = colBase0 + waveCol * 64 + sub * 16 + lmod;
            float badd = bias[ncol];
#pragma unroll
            for (int jj = 0; jj < 8; ++jj) {
                int r = waveRow * 32 + rt * 16 + jj + 8 * lhi;
                out[(b * S + tBase + r) * D + ncol] = acc[rt][sub][jj] + badd;
            }
        }
    }
}

extern "C" void kernel_launch(void* const* d_in, const int* in_sizes, int n_in,
                              void* d_out, int out_size, void* d_ws, size_t ws_size,
                              hipStream_t stream) {
    const float* x     = (const float*)d_in[0];
    const float* comp  = (const float*)d_in[1];
    const float* basis = (const float*)d_in[2];
    const float* root  = (const float*)d_in[3];
    const float* bias  = (const float*)d_in[4];
    float* out = (float*)d_out;

    float* wsf     = (float*)d_ws;
    float* sums    = wsf;                       // BATCH*D   = 2048 f
    float* coef    = wsf + BATCH * D;           // S*12      = 24576 f
    float* partial = coef + S * 12;             // BATCH*8*D = 16384 f

    colsum_partial<<<BATCH * 8, D, 0, stream>>>(x, partial);
    reduce_and_coef<<<S / 256, 256, 0, stream>>>(partial, comp, sums, coef);
    rgcn_gemm<<<dim3((BATCH * S) / BM, D / BN), 256, 0, stream>>>(
        x, basis, root, bias, sums, coef, out);
}